// BSplineActivation_36395552866878
// MI455X (gfx1250) — compile-verified
//
#include <hip/hip_runtime.h>
#include <stdint.h>

// ---------------------------------------------------------------------------
// B-spline activation (degree 3, 16 control points, open uniform knots).
// Memory-bound streaming kernel:
//   64 MiB traffic @ 23.3 TB/s  => ~2.9us floor.
// Per element: xs=clip(x/6+0.5,0,1); span k=floor(13*xs); t=13*xs-k;
//   y = ((c3*t+c2)*t+c1)*t+c0   with per-span cubic coeffs precomputed in LDS.
// Input streamed with the CDNA5 Tensor Data Mover (tensor_load_to_lds),
// double-buffered, synced with s_wait_tensorcnt + workgroup barriers.
// ---------------------------------------------------------------------------

#define TILE    4096        // floats per TDM tile (16 KB)
#define THREADS 256         // 8 wave32 per block

typedef unsigned int u32x4 __attribute__((ext_vector_type(4)));
typedef int          i32x4 __attribute__((ext_vector_type(4)));
typedef int          i32x8 __attribute__((ext_vector_type(8)));
typedef float        f32x4 __attribute__((ext_vector_type(4)));

__device__ __forceinline__ int clamp13(int v) {
    v = v < 0 ? 0 : v;
    return v > 13 ? 13 : v;
}

// Issue a 1-D TDM load: nelem f32 from gaddr -> LDS byte offset lds_off.
// D# layout per CDNA5 ISA ch.8 (group0: count/lds/global/type=2;
// group1: data_size=4B, tensor_dim0=nelem (OOB -> zero-fill), tile_dim0=nelem).
__device__ __forceinline__ void tdm_load_1d(uint64_t gaddr, uint32_t lds_off,
                                            int nelem) {
    u32x4 g0;
    g0[0] = 1u;                                    // count=1, user desc
    g0[1] = lds_off;                               // [63:32] lds_addr
    g0[2] = (uint32_t)gaddr;                       // global_addr[31:0]
    g0[3] = (uint32_t)((gaddr >> 32) & 0x01FFFFFFu)// global_addr[56:32]
          | 0x80000000u;                           // type=2 ("image")
    i32x8 g1;
    g1[0] = (2 << 16);                             // wg_mask=0, data_size=4B
    g1[1] = (int)(((uint32_t)nelem & 0xFFFFu) << 16);   // tensor_dim0[15:0]
    g1[2] = (int)(((uint32_t)nelem >> 16) & 0xFFFFu);   // tensor_dim0[31:16]
    g1[3] = (int)(((uint32_t)nelem & 0xFFFFu) << 16);   // tile_dim0
    g1[4] = 0;                                     // tile_dim1/2 = 0 (1-D)
    g1[5] = nelem;                                 // dim0 stride (benign)
    g1[6] = 0;
    g1[7] = 0;
    i32x4 gz = {0, 0, 0, 0};
#if defined(__clang_major__) && (__clang_major__ >= 23)
    i32x8 gz8 = {0, 0, 0, 0, 0, 0, 0, 0};
    __builtin_amdgcn_tensor_load_to_lds(g0, g1, gz, gz, gz8, 0);
#else
    __builtin_amdgcn_tensor_load_to_lds(g0, g1, gz, gz, 0);
#endif
}

__device__ __forceinline__ float bspline_eval(float xin, const f32x4* cf) {
    float xs  = fminf(fmaxf(fmaf(xin, 1.0f / 6.0f, 0.5f), 0.0f), 1.0f);
    float t13 = xs * 13.0f;
    int   k   = (int)t13;
    k = k > 12 ? 12 : k;
    float t = t13 - (float)k;
    f32x4 c = cf[k];                               // ds_load_b128, 16B stride:
                                                   // 13 spans -> disjoint banks
    return fmaf(fmaf(fmaf(c.w, t, c.z), t, c.y), t, c.x);
}

__global__ __launch_bounds__(THREADS) void
BSplineActivation_36395552866878_kernel(const float* __restrict__ x,
                                        const float* __restrict__ cp,
                                        float* __restrict__ out,
                                        int n, int tiles) {
    __shared__ __align__(16) float buf[2][TILE];
    __shared__ f32x4 s_coef[13];

    // ---- per-span cubic coefficients in local parameter t (de Boor on
    //      polynomials; denominators fold to small-integer reciprocals) ----
    if (threadIdx.x < 13) {
        const int k = threadIdx.x;
        float N[4][4] = {{0.f}};
        N[0][0] = 1.0f;
        for (int d = 1; d <= 3; ++d) {
            float sv0 = 0.f, sv1 = 0.f, sv2 = 0.f, sv3 = 0.f;
            for (int r = 0; r < d; ++r) {
                int   du  = clamp13(k + r + 1) - clamp13(k + 1 - d + r);
                float inv = 1.0f / (float)du;
                float R   = (float)(clamp13(k + r + 1) - k);   // right[r+1]=R-t
                float L   = (float)(k - clamp13(k + 1 - (d - r))); // left=t+L
                float t0 = N[r][0] * inv, t1 = N[r][1] * inv;
                float t2 = N[r][2] * inv, t3 = N[r][3] * inv;
                N[r][0] = sv0 + R * t0;
                N[r][1] = sv1 + R * t1 - t0;
                N[r][2] = sv2 + R * t2 - t1;
                N[r][3] = sv3 + R * t3 - t2;
                sv0 = L * t0;
                sv1 = L * t1 + t0;
                sv2 = L * t2 + t1;
                sv3 = L * t3 + t2;
            }
            N[d][0] = sv0; N[d][1] = sv1; N[d][2] = sv2; N[d][3] = sv3;
        }
        float c0 = 0.f, c1 = 0.f, c2 = 0.f, c3 = 0.f;
        for (int r = 0; r < 4; ++r) {
            float w = cp[k + r];
            c0 += N[r][0] * w; c1 += N[r][1] * w;
            c2 += N[r][2] * w; c3 += N[r][3] * w;
        }
        f32x4 cc = {c0, c1, c2, c3};
        s_coef[k] = cc;
    }

    // Variable-index touch so the compiler can't fold never-stored LDS reads.
    buf[0][threadIdx.x] = 0.0f;
    buf[1][threadIdx.x] = 0.0f;
    __syncthreads();   // init + coeff table complete before first TDM write

    const bool issuer = (threadIdx.x < 32);        // wave 0 (wave32)
    const uint32_t lds0 = (uint32_t)(uintptr_t)&buf[0][0];
    const uint32_t lds1 = (uint32_t)(uintptr_t)&buf[1][0];

    // Prologue: load first tile into buf0.
    if (issuer) {
        int64_t b0  = (int64_t)blockIdx.x * TILE;
        int     rem = (n - b0 > (int64_t)TILE) ? TILE : (int)(n - b0);
        tdm_load_1d((uint64_t)(uintptr_t)(x + b0), lds0, rem);
    }

    int p = 0;
    for (int tile = blockIdx.x; tile < tiles; tile += (int)gridDim.x) {
        if (issuer) __builtin_amdgcn_s_wait_tensorcnt(0); // tile data landed
        __syncthreads();                                  // visible to all waves

        // Prefetch next tile into the other buffer (overlaps with compute).
        int nextTile = tile + (int)gridDim.x;
        if (issuer && nextTile < tiles) {
            int64_t nb  = (int64_t)nextTile * TILE;
            int     rem = (n - nb > (int64_t)TILE) ? TILE : (int)(n - nb);
            tdm_load_1d((uint64_t)(uintptr_t)(x + nb), p ? lds0 : lds1, rem);
        }

        const f32x4* src  = (const f32x4*)&buf[p][0];
        int64_t      base = (int64_t)tile * TILE;
#pragma unroll
        for (int v = 0; v < TILE / (THREADS * 4); ++v) {
            int   e4 = v * THREADS + (int)threadIdx.x;    // coalesced b128
            f32x4 xv = src[e4];
            f32x4 yv;
            yv.x = bspline_eval(xv.x, s_coef);
            yv.y = bspline_eval(xv.y, s_coef);
            yv.z = bspline_eval(xv.z, s_coef);
            yv.w = bspline_eval(xv.w, s_coef);
            int64_t o = base + (int64_t)e4 * 4;
            if (o + 3 < (int64_t)n) {
                __builtin_nontemporal_store(yv, (f32x4*)(out + o)); // NT b128
            } else {
                float ys[4] = {yv.x, yv.y, yv.z, yv.w};
                for (int q = 0; q < 4; ++q)
                    if (o + q < (int64_t)n) out[o + q] = ys[q];
            }
        }
        __syncthreads();   // all waves done with buf[p] before it is refilled
        p ^= 1;
    }
}

extern "C" void kernel_launch(void* const* d_in, const int* in_sizes, int n_in,
                              void* d_out, int out_size, void* d_ws,
                              size_t ws_size, hipStream_t stream) {
    const float* x  = (const float*)d_in[0];
    const float* cp = (const float*)d_in[1];
    float*       out = (float*)d_out;
    const int    n   = in_sizes[0];
    if (n <= 0) return;
    int tiles  = (n + TILE - 1) / TILE;
    int blocks = tiles < 1024 ? tiles : 1024;
    hipLaunchKernelGGL(BSplineActivation_36395552866878_kernel,
                       dim3(blocks), dim3(THREADS), 0, stream,
                       x, cp, out, n, tiles);
}